// PredictionHead_64836826300448
// MI455X (gfx1250) — compile-verified
//
#include <hip/hip_runtime.h>
#include <hip/hip_bf16.h>

#define S_LEN 4096
#define NT    256
#define PER   (S_LEN / NT)   // 16 elements per thread

#define AS1 __attribute__((address_space(1)))
#define AS3 __attribute__((address_space(3)))

typedef int v4i_t __attribute__((__vector_size__(16)));

__global__ __launch_bounds__(NT)
void span_argmax_kernel(const float* __restrict__ start_logits,
                        const float* __restrict__ end_logits,
                        const int*   __restrict__ limit_ptr,
                        float*       __restrict__ out,
                        int B)
{
    __shared__ __align__(16) float s_sh[S_LEN];   // 16 KB
    __shared__ __align__(16) float e_sh[S_LEN];   // 16 KB
    __shared__ float red_v[NT];
    __shared__ int   red_i[NT];

    const int b = blockIdx.x;
    const int t = threadIdx.x;
    const int limit = *limit_ptr;

    const float* gs = start_logits + (size_t)b * S_LEN;
    const float* ge = end_logits   + (size_t)b * S_LEN;

    // ---- Stage rows into LDS via CDNA5 async global->LDS (ASYNCcnt path) ----
#if defined(__gfx1250__) && __has_builtin(__builtin_amdgcn_global_load_async_to_lds_b128)
    #pragma unroll
    for (int k = 0; k < PER / 4; ++k) {           // 4 x b128 per array per thread
        const int f4 = t + k * NT;                // float4 index, fully coalesced
        __builtin_amdgcn_global_load_async_to_lds_b128(
            (AS1 v4i_t*)(gs + 4 * f4),
            (AS3 v4i_t*)(&s_sh[4 * f4]), 0, 0);
        __builtin_amdgcn_global_load_async_to_lds_b128(
            (AS1 v4i_t*)(ge + 4 * f4),
            (AS3 v4i_t*)(&e_sh[4 * f4]), 0, 0);
    }
  #if __has_builtin(__builtin_amdgcn_s_wait_asynccnt)
    __builtin_amdgcn_s_wait_asynccnt(0);
  #else
    asm volatile("s_wait_asynccnt 0" ::: "memory");
  #endif
#else
    #pragma unroll
    for (int k = 0; k < PER / 4; ++k) {
        const int f4 = t + k * NT;
        ((float4*)s_sh)[f4] = ((const float4*)gs)[f4];
        ((float4*)e_sh)[f4] = ((const float4*)ge)[f4];
    }
#endif
    __syncthreads();

    // ---- start_pointer: argmax_i ( s[i] + max_{j in [i, i+limit]} e[j] ) ----
    float bestV = -__builtin_inff();
    int   bestI = 0x7fffffff;
    for (int k = 0; k < PER; ++k) {
        const int i = t + k * NT;
        int jmax = i + limit; if (jmax > S_LEN - 1) jmax = S_LEN - 1;
        float wm = e_sh[i];
        for (int j = i + 1; j <= jmax; ++j) wm = fmaxf(wm, e_sh[j]);
        const float v = s_sh[i] + wm;
        if (v > bestV || (v == bestV && i < bestI)) { bestV = v; bestI = i; }
    }
    red_v[t] = bestV; red_i[t] = bestI;
    __syncthreads();
    #pragma unroll
    for (int off = NT / 2; off > 0; off >>= 1) {
        if (t < off) {
            const float v2 = red_v[t + off];
            const int   i2 = red_i[t + off];
            if (v2 > red_v[t] || (v2 == red_v[t] && i2 < red_i[t])) {
                red_v[t] = v2; red_i[t] = i2;
            }
        }
        __syncthreads();
    }
    if (t == 0) out[b] = (float)red_i[0];
    __syncthreads();

    // ---- end_pointer: argmax_j ( e[j] + max_{i in [j-limit, j]} s[i] ) ----
    bestV = -__builtin_inff();
    bestI = 0x7fffffff;
    for (int k = 0; k < PER; ++k) {
        const int j = t + k * NT;
        int imin = j - limit; if (imin < 0) imin = 0;
        float wm = s_sh[imin];
        for (int i = imin + 1; i <= j; ++i) wm = fmaxf(wm, s_sh[i]);
        const float v = e_sh[j] + wm;
        if (v > bestV || (v == bestV && j < bestI)) { bestV = v; bestI = j; }
    }
    red_v[t] = bestV; red_i[t] = bestI;
    __syncthreads();
    #pragma unroll
    for (int off = NT / 2; off > 0; off >>= 1) {
        if (t < off) {
            const float v2 = red_v[t + off];
            const int   i2 = red_i[t + off];
            if (v2 > red_v[t] || (v2 == red_v[t] && i2 < red_i[t])) {
                red_v[t] = v2; red_i[t] = i2;
            }
        }
        __syncthreads();
    }
    if (t == 0) out[B + b] = (float)red_i[0];
}

extern "C" void kernel_launch(void* const* d_in, const int* in_sizes, int n_in,
                              void* d_out, int out_size, void* d_ws, size_t ws_size,
                              hipStream_t stream) {
    const float* start_logits = (const float*)d_in[0];
    const float* end_logits   = (const float*)d_in[1];
    const int*   answer_limit = (const int*)d_in[2];
    float* out = (float*)d_out;

    const int B = in_sizes[0] / S_LEN;   // 16

    span_argmax_kernel<<<dim3(B), dim3(NT), 0, stream>>>(
        start_logits, end_logits, answer_limit, out, B);
}